// RefAtomFeatureEmbedder_24412594111345
// MI455X (gfx1250) — compile-verified
//
#include <hip/hip_runtime.h>
#include <hip/hip_bf16.h>
#include <math.h>

typedef __attribute__((ext_vector_type(2))) float v2f;
typedef __attribute__((ext_vector_type(4))) float v4f;
typedef __attribute__((ext_vector_type(8))) float v8f;

#define BN_TOTAL (2 * 8192)        // B*N atom rows
#define K_FEAT   389               // 3 + 1 + 1 + 128 + 256
#define K_PAD    392               // padded to multiple of 4
#define K_CHUNKS (K_PAD / 4)       // 98
#define LDS_STRIDE 393             // conflict-free (393 % 64 = 9, coprime with 64)
#define C_ATOM   128
#define C_PAIR   16
#define CL_ELEMS (BN_TOTAL * C_ATOM)

// ---------------------------------------------------------------------------
// Kernel A: pack the combined weight matrix Bmat[k][n] = W_combined^T into ws.
// Rows 0..2: W_pos, 3: W_charge, 4: W_mask, 5..132: W_element, 133..388:
// W_chars, 389..391: zero pad.  Shape (K_PAD x 128), row-major.
// ---------------------------------------------------------------------------
__global__ void pack_weights(const float* __restrict__ W_pos,
                             const float* __restrict__ W_charge,
                             const float* __restrict__ W_mask,
                             const float* __restrict__ W_element,
                             const float* __restrict__ W_chars,
                             float* __restrict__ Bmat) {
    int i = blockIdx.x * blockDim.x + threadIdx.x;
    if (i >= K_PAD * C_ATOM) return;
    int k = i / C_ATOM;
    int n = i - k * C_ATOM;
    float v;
    if (k < 3)        v = W_pos[n * 3 + k];
    else if (k == 3)  v = W_charge[n];
    else if (k == 4)  v = W_mask[n];
    else if (k < 133) v = W_element[n * 128 + (k - 5)];
    else if (k < 389) v = W_chars[n * 256 + (k - 133)];
    else              v = 0.0f;
    Bmat[k * C_ATOM + n] = v;
}

// ---------------------------------------------------------------------------
// Kernel B: cl = X(16384 x 389) @ W^T(389 x 128) + bias, via f32 WMMA 16x16x4.
// Block = 256 threads (8 waves). Block owns a 16-row M tile; wave w owns the
// 16-column N stripe [16w, 16w+16). A-tile staged in LDS, B read from the
// packed matrix in ws (hot in L2 across all blocks).
// ---------------------------------------------------------------------------
__global__ void __launch_bounds__(256)
cl_gemm(const float* __restrict__ pos,
        const float* __restrict__ charge,
        const float* __restrict__ mask,
        const float* __restrict__ elem,
        const float* __restrict__ chars,
        const float* __restrict__ Bmat,
        const float* __restrict__ b_pos,
        const float* __restrict__ b_charge,
        const float* __restrict__ b_mask,
        const float* __restrict__ b_element,
        const float* __restrict__ b_chars,
        float* __restrict__ cl_out) {
    __shared__ float Xs[16 * LDS_STRIDE];

    const int tid = threadIdx.x;
    const int m0  = blockIdx.x * 16;

    // --- cooperative stage of the 16 x 392 A-tile (concatenated features) ---
    for (int i = tid; i < 16 * 128; i += 256) {          // element
        int r = i >> 7, j = i & 127;
        Xs[r * LDS_STRIDE + 5 + j] = elem[(m0 + r) * 128 + j];
    }
    for (int i = tid; i < 16 * 256; i += 256) {          // atom-name chars
        int r = i >> 8, j = i & 255;
        Xs[r * LDS_STRIDE + 133 + j] = chars[(m0 + r) * 256 + j];
    }
    if (tid < 48) {                                      // positions
        int r = tid / 3, kk = tid - r * 3;
        Xs[r * LDS_STRIDE + kk] = pos[(m0 + r) * 3 + kk];
    }
    if (tid < 16) {                                      // asinh(charge), mask, pad
        float x = charge[m0 + tid];
        float t = fabsf(x);
        float as = copysignf(logf(t + sqrtf(fmaf(t, t, 1.0f))), x);
        Xs[tid * LDS_STRIDE + 3] = as;
        Xs[tid * LDS_STRIDE + 4] = mask[m0 + tid];
        Xs[tid * LDS_STRIDE + 389] = 0.0f;
        Xs[tid * LDS_STRIDE + 390] = 0.0f;
        Xs[tid * LDS_STRIDE + 391] = 0.0f;
    }
    __syncthreads();

    // --- WMMA loop: one 16x16 output tile per wave -------------------------
    const int wave = tid >> 5;
    const int lane = tid & 31;
    const int half = lane >> 4;       // 0: lanes 0-15, 1: lanes 16-31
    const int mr   = lane & 15;       // A row / D column-lane
    const int n    = wave * 16 + mr;  // output column owned by this lane

    const float* Arow = &Xs[mr * LDS_STRIDE + 2 * half];
    const float* Bcol = Bmat + (2 * half) * C_ATOM + n;

    v8f c = {0.f, 0.f, 0.f, 0.f, 0.f, 0.f, 0.f, 0.f};

    for (int kc = 0; kc < K_CHUNKS; ++kc) {
        // A 16x4 f32 fragment: VGPR v holds K = 4*kc + 2*half + v
        v2f a;
        a.x = Arow[kc * 4];
        a.y = Arow[kc * 4 + 1];
        // B 4x16 f32 fragment: VGPR v holds K = 4*kc + 2*half + v, N = mr
        v2f b;
        b.x = Bcol[(kc * 4) * C_ATOM];
        b.y = Bcol[(kc * 4 + 1) * C_ATOM];
        c = __builtin_amdgcn_wmma_f32_16x16x4_f32(
                /*neg_a=*/false, a, /*neg_b=*/false, b,
                /*c_mod=*/(short)0, c, /*reuse_a=*/false, /*reuse_b=*/false);
    }

    const float bsum = b_pos[n] + b_charge[n] + b_mask[n] + b_element[n] + b_chars[n];

    // C/D layout: VGPR r -> M = r + 8*half, N = mr
    #pragma unroll
    for (int r = 0; r < 8; ++r) {
        int row = m0 + r + 8 * half;
        cl_out[row * C_ATOM + n] = c[r] + bsum;
    }
}

// ---------------------------------------------------------------------------
// Kernel C: pair features plm (2 x 256 x 32 x 128 x 16).  One thread per
// (b, nb, q, k) pair -> 16 contiguous f32 outputs (4x nontemporal v4f).
// ---------------------------------------------------------------------------
__global__ void __launch_bounds__(256)
pair_kernel(const float* __restrict__ pos,
            const int*   __restrict__ uid,
            const float* __restrict__ atom_mask,
            const float* __restrict__ W_offset,   // (16,3)
            const float* __restrict__ b_offset,
            const float* __restrict__ W_invd,     // (16,1)
            const float* __restrict__ b_invd,
            const float* __restrict__ W_valid,    // (16,1)
            const float* __restrict__ b_valid,
            float* __restrict__ plm_out) {
    __shared__ float sWo[48], sWi[16], sWv[16], sBo[16], sBi[16], sBv[16];
    const int t = threadIdx.x;
    if (t < 48) sWo[t] = W_offset[t];
    if (t < 16) {
        sWi[t] = W_invd[t];
        sWv[t] = W_valid[t];
        sBo[t] = b_offset[t];
        sBi[t] = b_invd[t];
        sBv[t] = b_valid[t];
    }
    __syncthreads();

    const int p = blockIdx.x * blockDim.x + threadIdx.x;   // 0 .. 2^21-1
    const int k  = p & 127;
    const int q  = (p >> 7) & 31;
    const int nb = (p >> 12) & 255;
    const int b  = p >> 20;

    const int qa   = nb * 32 + q;                 // query atom within N
    const int kidx = nb * 32 + k - 48;            // key atom (center-relative)
    const bool valid = (kidx >= 0) && (kidx < 8192);
    const int kc = valid ? kidx : (kidx < 0 ? 0 : 8191);

    const int qg = b * 8192 + qa;
    const int kg = b * 8192 + kc;

    const float qm = atom_mask[qg];
    const float km = atom_mask[kg] * (valid ? 1.0f : 0.0f);
    const float m  = qm * km;

    const float dx = (pos[qg * 3 + 0] - pos[kg * 3 + 0]) * m;
    const float dy = (pos[qg * 3 + 1] - pos[kg * 3 + 1]) * m;
    const float dz = (pos[qg * 3 + 2] - pos[kg * 3 + 2]) * m;

    const float vlm = (uid[qg] == uid[kg] ? 1.0f : 0.0f) * m;
    const float inv = 1.0f / (1.0f + fmaf(dx, dx, fmaf(dy, dy, dz * dz)));

    float out[16];
    #pragma unroll
    for (int c = 0; c < 16; ++c) {
        float dot = fmaf(dx, sWo[c * 3 + 0],
                    fmaf(dy, sWo[c * 3 + 1], dz * sWo[c * 3 + 2]));
        float val = vlm * (dot + sBo[c])
                  + vlm * (fmaf(inv, sWi[c], sBi[c]))
                  + vlm * (fmaf(vlm, sWv[c], sBv[c]));
        out[c] = val;
    }

    v4f* dst = reinterpret_cast<v4f*>(plm_out + (size_t)p * 16);
    #pragma unroll
    for (int j = 0; j < 4; ++j) {
        v4f v4 = { out[j * 4 + 0], out[j * 4 + 1],
                   out[j * 4 + 2], out[j * 4 + 3] };
        __builtin_nontemporal_store(v4, dst + j);
    }
}

// ---------------------------------------------------------------------------
extern "C" void kernel_launch(void* const* d_in, const int* in_sizes, int n_in,
                              void* d_out, int out_size, void* d_ws, size_t ws_size,
                              hipStream_t stream) {
    (void)in_sizes; (void)n_in; (void)out_size; (void)ws_size;

    const float* ref_pos    = (const float*)d_in[0];
    const float* ref_charge = (const float*)d_in[1];
    const float* ref_mask   = (const float*)d_in[2];
    const float* ref_elem   = (const float*)d_in[3];
    const float* ref_chars  = (const float*)d_in[4];   // (B,N,4,64) == (B,N,256)
    const int*   ref_uid    = (const int*)d_in[5];
    const float* atom_mask  = (const float*)d_in[6];
    const float* W_pos      = (const float*)d_in[7];
    const float* b_pos      = (const float*)d_in[8];
    const float* W_charge   = (const float*)d_in[9];
    const float* b_charge   = (const float*)d_in[10];
    const float* W_mask     = (const float*)d_in[11];
    const float* b_mask     = (const float*)d_in[12];
    const float* W_element  = (const float*)d_in[13];
    const float* b_element  = (const float*)d_in[14];
    const float* W_chars    = (const float*)d_in[15];
    const float* b_chars    = (const float*)d_in[16];
    const float* W_offset   = (const float*)d_in[17];
    const float* b_offset   = (const float*)d_in[18];
    const float* W_invd     = (const float*)d_in[19];
    const float* b_invd     = (const float*)d_in[20];
    const float* W_valid    = (const float*)d_in[21];
    const float* b_valid    = (const float*)d_in[22];
    // n_query=32, n_key=128 fixed by the reference setup.

    float* out  = (float*)d_out;
    float* Bmat = (float*)d_ws;                     // 392*128 f32 = 200704 B

    // 1) pack combined weight matrix into workspace
    pack_weights<<<(K_PAD * C_ATOM + 255) / 256, 256, 0, stream>>>(
        W_pos, W_charge, W_mask, W_element, W_chars, Bmat);

    // 2) cl: fused-feature GEMM via f32 WMMA (1024 blocks x 8 waves)
    cl_gemm<<<BN_TOTAL / 16, 256, 0, stream>>>(
        ref_pos, ref_charge, ref_mask, ref_elem, ref_chars, Bmat,
        b_pos, b_charge, b_mask, b_element, b_chars, out);

    // 3) plm: 2^21 pairs, one thread each
    pair_kernel<<<(2 * 256 * 32 * 128) / 256, 256, 0, stream>>>(
        ref_pos, ref_uid, atom_mask,
        W_offset, b_offset, W_invd, b_invd, W_valid, b_valid,
        out + CL_ELEMS);
}